// HypWeightAgg_54786602827843
// MI455X (gfx1250) — compile-verified
//
#include <hip/hip_runtime.h>
#include <stdint.h>

#define D 128
#define ROWS_PER_BLOCK 16
#define KC 64                 // edge chunk (K) per WMMA sweep
#define NSTEP (KC / 4)        // WMMA steps per chunk
#define EPSF 1e-15f
#define DEN_EPSF 1e-10f

typedef float v2f __attribute__((ext_vector_type(2)));
typedef float v8f __attribute__((ext_vector_type(8)));

__device__ __forceinline__ float waveReduceSum(float v) {
#pragma unroll
  for (int off = 16; off > 0; off >>= 1) v += __shfl_xor(v, off, 32);
  return v;
}

// base + zext(u32) -> SADDR + voffset global_load_b32 form
__device__ __forceinline__ float ldg_off(const float* __restrict__ base,
                                         uint32_t boff) {
  return *(const float*)((const char*)base + boff);
}

// ---------------------------------------------------------------------------
// Phase 1: per-node conformal factors. One wave (32 lanes) per node.
//   u    = max(||x_i||, EPS)
//   s[i] = sinh(2u)/u        (gamma_i * x_hyp_i == s[i] * x_i)
//   g[i] = cosh(2u)          (== gamma_i - 1)
// ---------------------------------------------------------------------------
__global__ __launch_bounds__(256) void node_factors(
    const float* __restrict__ x, float* __restrict__ s, float* __restrict__ g,
    int nNodes) {
  const int wave = threadIdx.x >> 5;
  const int lane = threadIdx.x & 31;
  const int node = blockIdx.x * (blockDim.x >> 5) + wave;
  if (node >= nNodes) return;

  const float4 xv = ((const float4*)(x + (size_t)node * D))[lane];
  float ss = xv.x * xv.x + xv.y * xv.y + xv.z * xv.z + xv.w * xv.w;
  ss = waveReduceSum(ss);
  const float u   = fmaxf(sqrtf(ss), EPSF);
  const float e2  = expf(2.0f * u);
  const float ie2 = 1.0f / e2;
  if (lane == 0) {
    s[node] = 0.5f * (e2 - ie2) / u;   // sinh(2u)/u
    g[node] = 0.5f * (e2 + ie2);       // cosh(2u)
  }
}

// ---------------------------------------------------------------------------
// Phase 2: segment-batched SpMM on the matrix pipes + fused finalize.
// Block = 256 threads = 8 wave32; 16 sorted destination rows per block.
// NUM[16x128] accumulated as sum_k A[16x4] x B[4x128] via V_WMMA_F32_16X16X4_F32.
// ---------------------------------------------------------------------------
__global__ __launch_bounds__(256) void spmm_wmma(
    const float* __restrict__ x, const int* __restrict__ rows,
    const int* __restrict__ cols, const float* __restrict__ vals,
    const float* __restrict__ s, const float* __restrict__ g,
    float* __restrict__ out, int nNodes, int nEdges) {
  __shared__ __align__(16) float    ldsA[KC][ROWS_PER_BLOCK]; // dense A panel
  __shared__ __align__(8)  uint32_t ldsOff[KC];               // col byte offset
  __shared__ __align__(8)  float    ldsS[KC];                 // sinh(2u)/u
  __shared__ float ldsVG[KC];                                 // val * (gamma-1)
  __shared__ int   ldsRowD[KC];                               // local dst row
  __shared__ float ldsNum[ROWS_PER_BLOCK][D];
  __shared__ float ldsDen[ROWS_PER_BLOCK];

  const int tid  = threadIdx.x;
  const int lane = tid & 31;
  const int wave = tid >> 5;                 // 0..7 : N-tile of 16 columns
  const int rowBase = blockIdx.x * ROWS_PER_BLOCK;

  // rows[] is sorted: binary-search this block's edge segment.
  auto lower = [&](int key) {
    int lo = 0, hi = nEdges;
    while (lo < hi) {
      int mid = (lo + hi) >> 1;
      if (rows[mid] < key) lo = mid + 1; else hi = mid;
    }
    return lo;
  };
  const int seg0 = lower(rowBase);
  const int seg1 = lower(rowBase + ROWS_PER_BLOCK);

  v8f acc = {};                        // 16x16 f32 accumulator tile (8 VGPRs)
  float den = 0.0f;                    // per-row denominator (threads 0..15)

  const int hsel  = lane >> 4;         // half-wave selects K pair (ISA layout)
  const int m     = lane & 15;         // A-matrix row / B column-in-tile
  const uint32_t nByte = (uint32_t)(wave * 16 + m) * 4u; // feature col (bytes)

  for (int base = seg0; base < seg1; base += KC) {
    const int kcur = min(KC, seg1 - base);
    __syncthreads();                   // LDS reuse across chunks
    if (tid < KC) {
      int      rloc = -1;
      uint32_t coff = 0;
      float    sv = 0.0f, vgv = 0.0f, vv = 0.0f;
      if (tid < kcur) {
        const int e = base + tid;
        const int c = cols[e];
        rloc = rows[e] - rowBase;      // 0..15 by segment construction
        coff = (uint32_t)c * (D * 4u);
        vv   = vals[e];
        sv   = s[c];
        vgv  = vv * g[c];
      }
      ldsRowD[tid] = rloc;
      ldsOff[tid]  = coff;
      ldsS[tid]    = sv;
      ldsVG[tid]   = vgv;
      // dense A row for this edge: zeros + val at its destination row
      const float4 z4 = {0.0f, 0.0f, 0.0f, 0.0f};
      float4* arow = (float4*)ldsA[tid];
      arow[0] = z4; arow[1] = z4; arow[2] = z4; arow[3] = z4;
      if (rloc >= 0) ldsA[tid][rloc] = vv;   // same-wave LDS ops are in order
    }
    if (base + KC < seg1 && tid < KC) {      // global_prefetch_b8 next chunk
      __builtin_prefetch(&cols[base + KC + tid], 0, 0);
      __builtin_prefetch(&vals[base + KC + tid], 0, 0);
    }
    __syncthreads();

    // -------- load burst: all B gathers of this chunk into registers -------
    float bx[NSTEP], by[NSTEP];
#pragma unroll
    for (int t = 0; t < NSTEP; ++t) {
      const int kb = 4 * t + 2 * hsel;
      const uint2 o2 = *(const uint2*)&ldsOff[kb];   // ds_load_b64
      bx[t] = ldg_off(x, o2.x + nByte);
      by[t] = ldg_off(x, o2.y + nByte);
    }
    // -------- WMMA burst ---------------------------------------------------
#pragma unroll
    for (int t = 0; t < NSTEP; ++t) {
      const int kb = 4 * t + 2 * hsel;
      v2f a;
      a.x = ldsA[kb][m];
      a.y = ldsA[kb + 1][m];
      const float2 s2 = *(const float2*)&ldsS[kb];   // ds_load_b64
      v2f b;
      b.x = s2.x * bx[t];
      b.y = s2.y * by[t];
      acc = __builtin_amdgcn_wmma_f32_16x16x4_f32(false, a, false, b,
                                                  (short)0, acc, false, false);
    }

    // Deterministic denominator: 16 serial scanners, fixed edge order.
    if (tid < ROWS_PER_BLOCK) {
#pragma unroll 4
      for (int k = 0; k < KC; ++k) {
        const float c = ldsVG[k];
        den += (ldsRowD[k] == tid) ? c : 0.0f;
      }
    }
  }

  // Spill the D-tile (ISA C/D layout: lanes 0-15 rows 0-7, lanes 16-31 rows 8-15).
  {
    const int nCol = wave * 16 + m;
#pragma unroll
    for (int i = 0; i < 8; ++i) ldsNum[i + 8 * hsel][nCol] = acc[i];
  }
  if (tid < ROWS_PER_BLOCK) ldsDen[tid] = den;
  __syncthreads();

  // Fused finalize: two_mean = num/den_clamped; out = 0.5*artanh(||v||)*v/||v||.
  // Wave w handles rows 2w, 2w+1; each lane owns 4 dims (float4 store).
#pragma unroll
  for (int rr = 0; rr < 2; ++rr) {
    const int r = wave * 2 + rr;
    const int grow = rowBase + r;
    const float dv   = ldsDen[r];
    const float dsgn = (dv >= 0.0f) ? 1.0f : -1.0f;
    const float inv  = 1.0f / (dsgn * fmaxf(fabsf(dv), DEN_EPSF));
    float4 tm;
    tm.x = ldsNum[r][lane * 4 + 0] * inv;
    tm.y = ldsNum[r][lane * 4 + 1] * inv;
    tm.z = ldsNum[r][lane * 4 + 2] * inv;
    tm.w = ldsNum[r][lane * 4 + 3] * inv;
    float ss = tm.x * tm.x + tm.y * tm.y + tm.z * tm.z + tm.w * tm.w;
    ss = waveReduceSum(ss);
    const float vn  = fmaxf(sqrtf(ss), EPSF);
    const float ain = fminf(vn, 1.0f - 1e-7f);
    const float at  = 0.5f * logf((1.0f + ain) / (1.0f - ain));  // artanh
    const float sc  = 0.5f * at / vn;
    if (grow < nNodes) {
      float4 o = {tm.x * sc, tm.y * sc, tm.z * sc, tm.w * sc};
      ((float4*)(out + (size_t)grow * D))[lane] = o;
    }
  }
}

// ---------------------------------------------------------------------------
extern "C" void kernel_launch(void* const* d_in, const int* in_sizes, int n_in,
                              void* d_out, int out_size, void* d_ws,
                              size_t ws_size, hipStream_t stream) {
  const float* x    = (const float*)d_in[0];
  const int*   rows = (const int*)d_in[1];
  const int*   cols = (const int*)d_in[2];
  const float* vals = (const float*)d_in[3];
  const int N = in_sizes[0] / D;
  const int E = in_sizes[1];

  float* s = (float*)d_ws;        // sinh(2u)/u   [N]
  float* g = s + N;               // cosh(2u)     [N]
  float* out = (float*)d_out;

  const int nb1 = (N + 7) / 8;    // 8 waves per block, one node per wave
  node_factors<<<nb1, 256, 0, stream>>>(x, s, g, N);

  const int nb2 = (N + ROWS_PER_BLOCK - 1) / ROWS_PER_BLOCK;
  spmm_wmma<<<nb2, 256, 0, stream>>>(x, rows, cols, vals, s, g, out, N, E);
}